// SequentialOmniMagnetTrajectoryField_31387620999596
// MI455X (gfx1250) — compile-verified
//
#include <hip/hip_runtime.h>
#include <math.h>

#ifndef __has_builtin
#define __has_builtin(x) 0
#endif

#if __has_builtin(__builtin_amdgcn_global_load_async_to_lds_b128)
#define OMNI_USE_ASYNC_LDS 1
#else
#define OMNI_USE_ASYNC_LDS 0
#endif

#if OMNI_USE_ASYNC_LDS
typedef int v4i_omni __attribute__((ext_vector_type(4)));
typedef __attribute__((address_space(1))) v4i_omni* omni_gptr;   // pointer to AS1 v4i
typedef __attribute__((address_space(3))) v4i_omni* omni_lptr;   // pointer to AS3 v4i
#endif

namespace {
constexpr int   kNWp            = 128;
constexpr float kEps            = 1e-12f;
constexpr float kMaxMoment      = 0.05f;
constexpr float kParticleMoment = 1e-14f;
constexpr float kDrag           = 1e-16f;
constexpr float kStaticFriction = 1e-19f;
constexpr float kMinDwell       = 0.01f;
constexpr float kMaxDwell       = 0.2f;
constexpr float kRelaxKeep      = 0.9048374180359596f;  // exp(-0.1)
constexpr float kEffR2          = 669.4330016f;         // (2*sqrt(3)*0.05/1e-5)^(2/3)
constexpr float kPi             = 3.14159265358979323846f;
}  // namespace

__global__ __launch_bounds__(256) void
omni_magnet_scan_kernel(const float* __restrict__ points,
                        const float* __restrict__ centers,
                        const float* __restrict__ raw_moment,
                        const float* __restrict__ raw_dwell,
                        float* __restrict__ out,
                        int n_points)
{
    // Raw staging: centers [0,384), raw_moment [384,768), raw_dwell [768,896)
    __shared__ __align__(16) float s_raw[kNWp * 7];
    // Processed waypoint table: 8 floats per waypoint {cx,cy,cz,mx,my,mz,dwell,pad}
    __shared__ __align__(16) float s_wp[kNWp * 8];

    const int tid = threadIdx.x;
    const int i = blockIdx.x * blockDim.x + tid;

    // ---- Issue the per-thread point load EARLY so its latency is hidden
    //      behind the async LDS staging + waypoint transform below.
    const int il = (i < n_points) ? i : (n_points - 1);
    const float px = points[3 * il + 0];
    const float py = points[3 * il + 1];
    const float pz = points[3 * il + 2];

    // ---- Stage waypoint data into LDS (gfx1250 async global->LDS path) ----
    if (tid < 224) {
        const float* src;
        int foff;
        if (tid < 96)       { src = centers;    foff = tid * 4; }
        else if (tid < 192) { src = raw_moment; foff = (tid - 96) * 4; }
        else                { src = raw_dwell;  foff = (tid - 192) * 4; }
        float* dst = &s_raw[tid * 4];
#if OMNI_USE_ASYNC_LDS
        __builtin_amdgcn_global_load_async_to_lds_b128(
            (omni_gptr)(src + foff),
            (omni_lptr)dst,
            0, 0);
#else
        *(float4*)dst = *(const float4*)(src + foff);
#endif
    }
#if OMNI_USE_ASYNC_LDS
#if __has_builtin(__builtin_amdgcn_s_wait_asynccnt)
    __builtin_amdgcn_s_wait_asynccnt(0);
#else
    asm volatile("s_wait_asynccnt 0" ::: "memory");
#endif
#endif
    __syncthreads();

    // ---- Transform raw waypoint params into the per-step table ----
    if (tid < kNWp) {
        const float cx = s_raw[3 * tid + 0];
        const float cy = s_raw[3 * tid + 1];
        const float cz = s_raw[3 * tid + 2];
        const float mx = kMaxMoment * tanhf(s_raw[384 + 3 * tid + 0]);
        const float my = kMaxMoment * tanhf(s_raw[384 + 3 * tid + 1]);
        const float mz = kMaxMoment * tanhf(s_raw[384 + 3 * tid + 2]);
        const float sg = 1.0f / (1.0f + expf(-s_raw[768 + tid]));
        const float dw = kMinDwell + (kMaxDwell - kMinDwell) * sg;
        float* w = &s_wp[tid * 8];
        w[0] = cx; w[1] = cy; w[2] = cz;
        w[3] = mx; w[4] = my; w[5] = mz;
        w[6] = dw; w[7] = 0.0f;
    }
    __syncthreads();

    if (i >= n_points) return;

    const float pxy2 = px * px + py * py;

    // ---- Per-point state, register resident for the entire 128-step scan ----
    float ox = 0.0f, oy = 0.0f, oz = 1.0f;
    float at = 0.0f, peak = 0.0f;

    for (int k = 0; k < kNWp; ++k) {
        const float* w = &s_wp[k * 8];
        const float cx = w[0], cy = w[1], cz = w[2];
        const float mx = w[3], my = w[4], mz = w[5];
        const float dt = w[6];

        // Dipole field at the point
        const float rx = px - cx, ry = py - cy, rz = pz - cz;
        const float r2 = fmaxf(rx * rx + ry * ry + rz * rz, kEps);
        const float r = sqrtf(r2);
        const float inv_r3 = 1.0f / (r2 * r);
        const float inv_r5 = inv_r3 / r2;
        const float mdotr = rx * mx + ry * my + rz * mz;
        const float s3 = 3.0f * (mdotr * inv_r5);
        const float fx = rx * s3 - mx * inv_r3;
        const float fy = ry * s3 - my * inv_r3;
        const float fz = rz * s3 - mz * inv_r3;

        const float bmag = sqrtf(fx * fx + fy * fy + fz * fz);
        const float invb = 1.0f / fmaxf(bmag, kEps);
        const float bx = fx * invb, by = fy * invb, bz = fz * invb;

        // Candidate check (xy disc)
        const float hr2 = fmaxf(kEffR2 - cz * cz, 0.0f);
        const float dxy2 = pxy2 + (cx * cx + cy * cy) - 2.0f * (px * cx + py * cy);
        const bool cand = dxy2 <= hr2;

        // Rotation toward field direction
        const float crx = oy * bz - oz * by;
        const float cry = oz * bx - ox * bz;
        const float crz = ox * by - oy * bx;
        const float axn = sqrtf(crx * crx + cry * cry + crz * crz);
        const float sin_t = fminf(fmaxf(axn, 0.0f), 1.0f);
        const float cos_t = fminf(fmaxf(ox * bx + oy * by + oz * bz, -1.0f), 1.0f);

        const bool active = (kParticleMoment * bmag * sin_t) > kStaticFriction;
        const float rate = (kParticleMoment * bmag) / kDrag;
        const float tan_half = sin_t / fmaxf(1.0f + cos_t, kEps);
        const float theta = atan2f(sin_t, cos_t);
        const float theta_new = 2.0f * atanf(tan_half * expf(-rate * dt));
        const float rot = active ? fminf(fmaxf(theta - theta_new, 0.0f), kPi) : 0.0f;

        const float inva = 1.0f / fmaxf(axn, kEps);
        const float ux = crx * inva, uy = cry * inva, uz = crz * inva;
        const float ca = cosf(rot);
        const float sa = sinf(rot);
        const float t1 = (ux * ox + uy * oy + uz * oz) * (1.0f - ca);
        const float vx = uy * oz - uz * oy;   // unit x orient
        const float vy = uz * ox - ux * oz;
        const float vz = ux * oy - uy * ox;
        const float nx = ox * ca + vx * sa + ux * t1;
        const float ny = oy * ca + vy * sa + uy * t1;
        const float nz = oz * ca + vz * sa + uz * t1;

        const bool apply = cand && (axn > 1e-6f);
        ox = apply ? nx : ox;
        oy = apply ? ny : oy;
        oz = apply ? nz : oz;

        at += (cand && active) ? dt : 0.0f;

        const float bxy = sqrtf(fx * fx + fy * fy);
        peak = fmaxf(peak * kRelaxKeep, cand ? bxy : 0.0f);
    }

    float* o = out + (size_t)i * 5;
    o[0] = ox; o[1] = oy; o[2] = oz; o[3] = at; o[4] = peak;
}

extern "C" void kernel_launch(void* const* d_in, const int* in_sizes, int n_in,
                              void* d_out, int out_size, void* d_ws, size_t ws_size,
                              hipStream_t stream) {
    (void)n_in; (void)out_size; (void)d_ws; (void)ws_size;
    const float* points     = (const float*)d_in[0];
    const float* centers    = (const float*)d_in[1];
    const float* raw_moment = (const float*)d_in[2];
    const float* raw_dwell  = (const float*)d_in[3];
    float* out = (float*)d_out;

    const int n_points = in_sizes[0] / 3;   // 65536
    const int block = 256;
    const int grid = (n_points + block - 1) / block;
    omni_magnet_scan_kernel<<<grid, block, 0, stream>>>(
        points, centers, raw_moment, raw_dwell, out, n_points);
}